// BinsCombinerLayer_8916352106657
// MI455X (gfx1250) — compile-verified
//
#include <hip/hip_runtime.h>

// out = (1/Nrows) * dot(probs, centroids) over all 101M elements.
// Memory-bound: 808 MB @ 23.3 TB/s ~= 35us floor. WMMA f32 16x16x4 is the
// (free) multiply-reduce engine via the trace trick; b128 non-temporal loads
// carry the bandwidth. Main loop is guard-free (full wave chunks only) so no
// exec masking / zero-init in the hot path; remainder runs on one wave.

typedef __attribute__((ext_vector_type(2))) float v2f;
typedef __attribute__((ext_vector_type(4))) float v4f;
typedef __attribute__((ext_vector_type(8))) float v8f;

__device__ __forceinline__ float wave_reduce(float v) {
#pragma unroll
  for (int off = 16; off > 0; off >>= 1) v += __shfl_xor(v, off, 32);
  return v;
}

__device__ __forceinline__ v8f wmma_acc(v4f px, v4f py, v8f acc) {
  // Same (lane,slot)->(m,k) mapping for A and B => diagonal accumulates x*y.
  v2f a0 = {px.x, px.y}, b0 = {py.x, py.y};
  v2f a1 = {px.z, px.w}, b1 = {py.z, py.w};
  acc = __builtin_amdgcn_wmma_f32_16x16x4_f32(
      false, a0, false, b0, (short)0, acc, false, false);
  acc = __builtin_amdgcn_wmma_f32_16x16x4_f32(
      false, a1, false, b1, (short)0, acc, false, false);
  return acc;
}

__global__ void bins_dot_wmma_kernel(const float* __restrict__ x,
                                     const float* __restrict__ y,
                                     float* __restrict__ partials,
                                     long long nvec4, long long total) {
  const unsigned lane = threadIdx.x & 31u;
  const unsigned wib  = threadIdx.x >> 5;  // wave index within block
  const long long wave_global =
      ((long long)blockIdx.x * blockDim.x + threadIdx.x) >> 5;
  const long long nwaves = ((long long)gridDim.x * blockDim.x) >> 5;

  const long long nchunks_full = nvec4 >> 5;   // chunks with all 32 lanes in-bounds
  const long long npairs = nchunks_full >> 1;  // unroll x2: 64 float4s per iter

  const v4f* __restrict__ x4 = (const v4f*)x;
  const v4f* __restrict__ y4 = (const v4f*)y;

  // D = A*B + C accumulator; only the diagonal is meaningful (trace trick).
  v8f acc = {};

  // Guard-free main loop: wave-uniform trip count, EXEC all-ones throughout,
  // 4 x global_load_b128 NT + 4 x v_wmma per iteration (1 KB/array/iter/wave).
  for (long long p = wave_global; p < npairs; p += nwaves) {
    const long long i0 = (p << 6) + lane;       // first chunk of the pair
    const long long i1 = i0 + 32;               // second chunk of the pair
    v4f px0 = __builtin_nontemporal_load(&x4[i0]);
    v4f py0 = __builtin_nontemporal_load(&y4[i0]);
    v4f px1 = __builtin_nontemporal_load(&x4[i1]);
    v4f py1 = __builtin_nontemporal_load(&y4[i1]);
    acc = wmma_acc(px0, py0, acc);
    acc = wmma_acc(px1, py1, acc);
  }

  // Diagonal extraction per C/D layout: VGPR r holds M=r (lanes 0-15, N=lane)
  // and M=8+r (lanes 16-31, N=lane-16). Diag: lane r -> acc[r]; lane 24+r -> acc[r].
  float dv[8] = {acc[0], acc[1], acc[2], acc[3], acc[4], acc[5], acc[6], acc[7]};
  float v = 0.f;
#pragma unroll
  for (int r = 0; r < 8; ++r) {
    v += (lane == (unsigned)r)        ? dv[r] : 0.f;
    v += (lane == (unsigned)(24 + r)) ? dv[r] : 0.f;
  }

  // Remainder (odd chunk + partial chunk + total%4), one wave, plain FMA —
  // no WMMA here so lane divergence is harmless. <64 float4s + <4 scalars.
  if (blockIdx.x == 0 && wib == 0) {
    for (long long i = (npairs << 6) + lane; i < nvec4; i += 32) {
      v4f px = x4[i], py = y4[i];
      v = fmaf(px.x, py.x, v);
      v = fmaf(px.y, py.y, v);
      v = fmaf(px.z, py.z, v);
      v = fmaf(px.w, py.w, v);
    }
    for (long long e = (nvec4 << 2) + lane; e < total; e += 32)
      v = fmaf(x[e], y[e], v);
  }

  // Wave + block reduction -> one partial per block (no global atomics).
  v = wave_reduce(v);
  __shared__ float sdata[32];
  if (lane == 0) sdata[wib] = v;
  __syncthreads();
  if (wib == 0) {
    const unsigned nw = blockDim.x >> 5;
    float bv = (lane < nw) ? sdata[lane] : 0.f;
    bv = wave_reduce(bv);
    if (lane == 0) partials[blockIdx.x] = bv;
  }
}

__global__ void bins_final_reduce_kernel(const float* __restrict__ partials,
                                         float* __restrict__ out,
                                         int nblocks, float scale) {
  float v = 0.f;
  for (int i = threadIdx.x; i < nblocks; i += blockDim.x) v += partials[i];
  v = wave_reduce(v);
  __shared__ float sdata[32];
  const unsigned lane = threadIdx.x & 31u, wib = threadIdx.x >> 5;
  if (lane == 0) sdata[wib] = v;
  __syncthreads();
  if (wib == 0) {
    float bv = (lane < (blockDim.x >> 5)) ? sdata[lane] : 0.f;
    bv = wave_reduce(bv);
    if (lane == 0) out[0] = bv * scale;
  }
}

extern "C" void kernel_launch(void* const* d_in, const int* in_sizes, int n_in,
                              void* d_out, int out_size, void* d_ws, size_t ws_size,
                              hipStream_t stream) {
  (void)n_in; (void)out_size;
  const float* probs = (const float*)d_in[0];
  const float* cents = (const float*)d_in[1];

  const long long total = (long long)in_sizes[0];  // N*K flat elements
  const int K = 101;                               // fixed by reference (ages 0..100)
  long long nrows = total / K;
  if (nrows <= 0) nrows = 1;
  const float scale = 1.0f / (float)nrows;
  const long long nvec4 = total >> 2;

  // 2048 blocks x 256 threads = 16384 waves; ~48 KB streamed per wave.
  int nblocks = 2048;
  const long long maxp = (long long)(ws_size / sizeof(float));
  if (maxp < nblocks) nblocks = (int)(maxp > 0 ? maxp : 1);
  float* partials = (float*)d_ws;

  bins_dot_wmma_kernel<<<nblocks, 256, 0, stream>>>(probs, cents, partials,
                                                    nvec4, total);
  bins_final_reduce_kernel<<<1, 256, 0, stream>>>(partials, (float*)d_out,
                                                  nblocks, scale);
}